// SE_block_73607149519238
// MI455X (gfx1250) — compile-verified
//
#include <hip/hip_runtime.h>
#include <hip/hip_bf16.h>

// Shapes fixed by the reference.
#define B_    32
#define M_    2048
#define D_    512
#define H_    64
#define WIN_  16
#define NW_   (M_ / WIN_)      // 128 windows
#define WGW_  16               // windows handled per workgroup (WMMA M dim)
#define GPB_  (NW_ / WGW_)     // 8 window-groups per batch
#define MPITCH 516             // D_ + 4 pad floats -> conflict-free WMMA A reads
#define HPITCH 68              // H_ + 4 pad floats

typedef float v2f __attribute__((ext_vector_type(2)));
typedef float v8f __attribute__((ext_vector_type(8)));

// Fast activations: denominators are >= 1, so plain v_rcp_f32 is safe.
__device__ __forceinline__ float softsign_fast(float v) {
    return v * __builtin_amdgcn_rcpf(1.f + fabsf(v));
}
__device__ __forceinline__ float sigmoid_fast(float v) {
    return __builtin_amdgcn_rcpf(1.f + __expf(-v));
}

__global__ __launch_bounds__(256) void se_block_kernel(
    const float* __restrict__ x,  const float* __restrict__ W1,
    const float* __restrict__ b1, const float* __restrict__ W2,
    const float* __restrict__ b2, float* __restrict__ y)
{
    __shared__ float ms[WGW_ * MPITCH];   // window means; reused for gate g
    __shared__ float hs[WGW_ * HPITCH];   // hidden activations

    const int t    = threadIdx.x;
    const int lane = t & 31;
    const int wave = t >> 5;
    const int b    = blockIdx.x / GPB_;
    const int wg   = blockIdx.x % GPB_;

    const float* xblk = x + ((size_t)b * M_ + (size_t)wg * (WGW_ * WIN_)) * D_;
    float*       yblk = y + ((size_t)b * M_ + (size_t)wg * (WGW_ * WIN_)) * D_;

    // ---------- Phase 1: windowed mean-pool -> LDS (coalesced float2) ------
    // Regular-temporal loads: this chunk is re-read in phase 3 (L2 hit).
    const int c0 = t * 2;                         // 256 threads * 2 cols = 512
    for (int w = 0; w < WGW_; ++w) {
        float s0 = 0.f, s1 = 0.f;
        const float* xr = xblk + (size_t)(w * WIN_) * D_ + c0;
        #pragma unroll
        for (int r = 0; r < WIN_; ++r) {
            v2f v = *(const v2f*)(xr + (size_t)r * D_);
            s0 += v.x; s1 += v.y;
        }
        ms[w * MPITCH + c0]     = s0 * (1.f / WIN_);
        ms[w * MPITCH + c0 + 1] = s1 * (1.f / WIN_);
    }
    __syncthreads();

    // ISA f32 16x16x4 operand mapping (wave32):
    //   A: lane L<16 -> row L, K = k+0/k+1 ; lane L>=16 -> row L-16, K = k+2/k+3
    //   B: same K split per half-wave, N = lane&15
    //   C/D: VGPR v -> row v (lanes 0-15) / row v+8 (lanes 16-31), col = lane&15
    const int mrow  = lane & 15;
    const int kk    = (lane >> 4) * 2;
    const int rbase = (lane >> 4) * 8;

    // ---------- Phase 2a: h = softsign(m @ W1 + b1) ------------------------
    if (wave < 4) {                               // 4 waves cover N = 64
        const int ntile = wave;
        const int ncol  = ntile * 16 + (lane & 15);
        const float* w1c = W1 + ncol;             // W1 is [D_][H_] row-major
        const float* mrp = ms + mrow * MPITCH + kk;
        v8f acc = {};
        #pragma unroll 4
        for (int k = 0; k < D_; k += 4) {
            v2f a;  a.x  = mrp[k];              a.y  = mrp[k + 1];
            v2f bb; bb.x = w1c[(k + kk) * H_];  bb.y = w1c[(k + kk + 1) * H_];
            acc = __builtin_amdgcn_wmma_f32_16x16x4_f32(
                false, a, false, bb, (short)0, acc, false, false);
        }
        const float bias = b1[ncol];
        #pragma unroll
        for (int v = 0; v < 8; ++v) {
            hs[(rbase + v) * HPITCH + ncol] = softsign_fast(acc[v] + bias);
        }
    }
    __syncthreads();

    // ---------- Phase 2b: g = sigmoid(softsign(h @ W2 + b2)) ---------------
    {
        const float* hrp = hs + mrow * HPITCH + kk;
        #pragma unroll
        for (int nt = 0; nt < 4; ++nt) {          // 8 waves * 4 tiles = N 512
            const int ntile = wave * 4 + nt;
            const int ncol  = ntile * 16 + (lane & 15);
            const float* w2c = W2 + ncol;         // W2 is [H_][D_] row-major
            v8f acc = {};
            #pragma unroll
            for (int k = 0; k < H_; k += 4) {
                v2f a;  a.x  = hrp[k];              a.y  = hrp[k + 1];
                v2f bb; bb.x = w2c[(k + kk) * D_];  bb.y = w2c[(k + kk + 1) * D_];
                acc = __builtin_amdgcn_wmma_f32_16x16x4_f32(
                    false, a, false, bb, (short)0, acc, false, false);
            }
            const float bias = b2[ncol];
            #pragma unroll
            for (int v = 0; v < 8; ++v) {
                float s = softsign_fast(acc[v] + bias);
                ms[(rbase + v) * MPITCH + ncol] = sigmoid_fast(s);
            }
        }
    }
    __syncthreads();

    // ---------- Phase 3: y = x * g ------------------------------------------
    // x re-read: last use -> non-temporal load (frees L2 as we go).
    // y: pure streaming output -> non-temporal store (don't pollute L2).
    for (int w = 0; w < WGW_; ++w) {
        v2f g = *(const v2f*)(ms + w * MPITCH + c0);
        const float* xr = xblk + (size_t)(w * WIN_) * D_ + c0;
        float*       yr = yblk + (size_t)(w * WIN_) * D_ + c0;
        #pragma unroll
        for (int r = 0; r < WIN_; ++r) {
            v2f v = __builtin_nontemporal_load((const v2f*)(xr + (size_t)r * D_));
            v2f o; o.x = v.x * g.x; o.y = v.y * g.y;
            __builtin_nontemporal_store(o, (v2f*)(yr + (size_t)r * D_));
        }
    }
}

extern "C" void kernel_launch(void* const* d_in, const int* in_sizes, int n_in,
                              void* d_out, int out_size, void* d_ws, size_t ws_size,
                              hipStream_t stream) {
    const float* x  = (const float*)d_in[0];
    const float* W1 = (const float*)d_in[1];
    const float* b1 = (const float*)d_in[2];
    const float* W2 = (const float*)d_in[3];
    const float* b2 = (const float*)d_in[4];
    float* y = (float*)d_out;
    (void)in_sizes; (void)n_in; (void)out_size; (void)d_ws; (void)ws_size;

    dim3 grid(B_ * GPB_);   // 256 workgroups
    dim3 block(256);        // 8 wave32s
    se_block_kernel<<<grid, block, 0, stream>>>(x, W1, b1, W2, b2, y);
}